// QGNN2_28217935135270
// MI455X (gfx1250) — compile-verified
//
#include <hip/hip_runtime.h>
#include <cstdint>

// ---------------------------------------------------------------------------
// GNN message-passing network on MI455X (gfx1250), fp32 end-to-end.
// Matrix math: V_WMMA_F32_16X16X4_F32, one 16-row tile per wave32.
// Weights pre-swizzled into B-fragment layout in LDS; activations staged in
// padded LDS tiles; segment_sum via global_atomic_add_f32.
// Bias is folded into the WMMA C-operand init (every acc register of a given
// (nt,lane) maps to one output column). SiLU uses hardware V_RCP_F32.
// ---------------------------------------------------------------------------

#define N_NODES 50000
#define N_EDGES 800000
#define WPB 4      // waves per block
#define TPB 128    // threads per block

typedef float v2f __attribute__((ext_vector_type(2)));
typedef float v8f __attribute__((ext_vector_type(8)));

__device__ inline v8f vsplat8(float s) {
  v8f z = {s, s, s, s, s, s, s, s};
  return z;
}

// silu(x) = x * sigmoid(x); V_EXP_F32 + V_RCP_F32 (~1ulp, matches f32 ref well)
__device__ inline float silu(float x) {
  float e = __expf(-x);
  return x * __builtin_amdgcn_rcpf(1.0f + e);
}

// D = A(16x4 f32) * B(4x16 f32) + C(16x16 f32)
__device__ inline v8f wmma4(v2f a, v2f b, v8f c) {
  return __builtin_amdgcn_wmma_f32_16x16x4_f32(
      /*neg_a=*/false, a, /*neg_b=*/false, b,
      /*c_mod=*/(short)0, c, /*reuse_a=*/false, /*reuse_b=*/false);
}

// Stage W[K, ncols] (row-major) into LDS in B-fragment order.
// Fragment (g = k0/4, nt = n0/16) holds 64 floats laid out so that lane L
// reads its v2f at offset frag*64 + L*2:
//   v0 lanes 0-15  = W[k0+0][n0+L]     v0 lanes 16-31 = W[k0+1][n0+L-16]
//   v1 lanes 0-15  = W[k0+2][n0+L]     v1 lanes 16-31 = W[k0+3][n0+L-16]
__device__ inline void stage_weights(const float* __restrict__ W, float* wf,
                                     int K, int ncols, int NT, int tid, int nth) {
  int total = K * NT * 16;
  for (int i = tid; i < total; i += nth) {
    int k = i / (NT * 16);
    int n = i - k * (NT * 16);
    float v = (n < ncols) ? W[k * ncols + n] : 0.0f;
    int g = k >> 2, r = k & 3, nt = n >> 4, c = n & 15;
    wf[(g * NT + nt) * 64 + ((c + ((r & 1) << 4)) << 1) + (r >> 1)] = v;
  }
}

__device__ inline void stage_bias(const float* __restrict__ B, float* b,
                                  int ncols, int tid, int nth) {
  for (int i = tid; i < 64; i += nth) b[i] = (i < ncols) ? B[i] : 0.0f;
}

// Init accumulators with the bias: C element (reg r, lane) has column
// n = nt*16 + lane%16, independent of r -> splat bias[n] into all 8 regs.
template <int NT>
__device__ inline void acc_bias_init(v8f acc[NT], const float* bias, int lane) {
  int c = lane & 15;
#pragma unroll
  for (int nt = 0; nt < NT; ++nt) acc[nt] = vsplat8(bias[nt * 16 + c]);
}

// [16,K]@[K,16*NT] from LDS state (row-major, padded stride rs) and
// swizzled LDS weight fragments. A-frag is one ds_load_b64 per lane.
template <int NT>
__device__ inline void gemm_lds(const float* st, int rs, const float* wf,
                                int K, int lane, v8f acc[NT]) {
  int m = lane & 15;
  int koff = (lane >> 4) << 1;  // lanes 16-31 hold K+2,K+3 of their row
  for (int k0 = 0; k0 < K; k0 += 4) {
    const float* pa = st + m * rs + k0 + koff;
    v2f a;
    a.x = pa[0];
    a.y = pa[1];
    int g = k0 >> 2;
#pragma unroll
    for (int nt = 0; nt < NT; ++nt) {
      const float* pb = wf + (g * NT + nt) * 64 + (lane << 1);
      v2f b;
      b.x = pb[0];
      b.y = pb[1];
      acc[nt] = wmma4(a, b, acc[nt]);
    }
  }
}

// C-layout (VGPR r: lanes0-15 -> M=r, lanes16-31 -> M=r+8; N = nt*16+lane%16)
// back to row-major LDS tile, with optional SiLU (bias already in acc).
template <int NT>
__device__ inline void c_to_lds(float* ht, int rs, int lane, const v8f acc[NT],
                                bool act) {
  int c = lane & 15;
  int mb = (lane >> 4) << 3;
#pragma unroll
  for (int nt = 0; nt < NT; ++nt) {
    int n = nt * 16 + c;
#pragma unroll
    for (int r = 0; r < 8; ++r) {
      float v = acc[nt][r];
      if (act) v = silu(v);
      ht[(mb + r) * rs + n] = v;
    }
  }
}

// ---------------------------------------------------------------------------
// Embed: [rows,16] -> Linear(16,64) -> SiLU -> Linear(64,64)   (plain end)
// ---------------------------------------------------------------------------
__global__ void k_embed(const float* __restrict__ X, int ntiles,
                        const float* __restrict__ W1, const float* __restrict__ B1,
                        const float* __restrict__ W2, const float* __restrict__ B2,
                        float* __restrict__ Y) {
  extern __shared__ float smem[];
  const int K1 = 16, RS1 = 20, RSH = 68;
  float* wf1 = smem;                 // 16*64
  float* wf2 = wf1 + K1 * 64;        // 64*64
  float* b1 = wf2 + 64 * 64;
  float* b2 = b1 + 64;
  float* wbase = b2 + 64;
  int tid = threadIdx.x, lane = tid & 31, wid = tid >> 5;
  stage_weights(W1, wf1, K1, 64, 4, tid, TPB);
  stage_weights(W2, wf2, 64, 64, 4, tid, TPB);
  stage_bias(B1, b1, 64, tid, TPB);
  stage_bias(B2, b2, 64, tid, TPB);
  __syncthreads();
  float* st = wbase + wid * (16 * RS1 + 16 * RSH);
  float* ht = st + 16 * RS1;
  int gw = blockIdx.x * WPB + wid, nw = gridDim.x * WPB;
  for (int tile = gw; tile < ntiles; tile += nw) {
    int r0 = tile * 16;
    for (int i = lane; i < 64; i += 32) {  // 16 rows x 4 float4
      int row = i >> 2, c4 = i & 3;
      float4 v = *(const float4*)(X + (size_t)(r0 + row) * 16 + (c4 << 2));
      *(float4*)(st + row * RS1 + (c4 << 2)) = v;
    }
    v8f acc[4];
    acc_bias_init<4>(acc, b1, lane);
    gemm_lds<4>(st, RS1, wf1, K1, lane, acc);
    c_to_lds<4>(ht, RSH, lane, acc, true);
    v8f acc2[4];
    acc_bias_init<4>(acc2, b2, lane);
    gemm_lds<4>(ht, RSH, wf2, 64, lane, acc2);
    int c = lane & 15, mb = (lane >> 4) << 3;
#pragma unroll
    for (int nt = 0; nt < 4; ++nt) {
      int n = nt * 16 + c;
#pragma unroll
      for (int r = 0; r < 8; ++r)
        Y[(size_t)(r0 + mb + r) * 64 + n] = acc2[nt][r];
    }
  }
}

// ---------------------------------------------------------------------------
// Edge pass: state=[h_n[s]|h_n[r]|h_e] (192) -> L(192,64)->SiLU->L(64,64)->SiLU
// writes new h_e, atomically accumulates into aggr[receiver].
// ---------------------------------------------------------------------------
__global__ void k_edge_layer(const float* __restrict__ HN, float* __restrict__ HE,
                             const int* __restrict__ EIDX, float* __restrict__ AG,
                             const float* __restrict__ W1, const float* __restrict__ B1,
                             const float* __restrict__ W2, const float* __restrict__ B2) {
  extern __shared__ float smem[];
  const int K1 = 192, RS1 = 196, RSH = 68;
  float* wf1 = smem;                 // 192*64
  float* wf2 = wf1 + K1 * 64;        // 64*64
  float* b1 = wf2 + 64 * 64;
  float* b2 = b1 + 64;
  float* wbase = b2 + 64;
  int tid = threadIdx.x, lane = tid & 31, wid = tid >> 5;
  stage_weights(W1, wf1, K1, 64, 4, tid, TPB);
  stage_weights(W2, wf2, 64, 64, 4, tid, TPB);
  stage_bias(B1, b1, 64, tid, TPB);
  stage_bias(B2, b2, 64, tid, TPB);
  __syncthreads();
  float* st = wbase + wid * (16 * RS1 + 16 * RSH + 32);
  float* ht = st + 16 * RS1;
  int* iarr = (int*)(ht + 16 * RSH);  // [0..15]=sender, [16..31]=receiver
  int gw = blockIdx.x * WPB + wid, nw = gridDim.x * WPB;
  const int ntiles = N_EDGES / 16;
  for (int tile = gw; tile < ntiles; tile += nw) {
    int e0 = tile * 16;
    {
      int m = lane & 15;
      iarr[lane] = (lane < 16) ? EIDX[e0 + m] : EIDX[N_EDGES + e0 + m];
    }
    // gather 16 x 192 state tile as float4s (768 total, 24 per lane)
    for (int i = lane; i < 768; i += 32) {
      int row = i / 48;
      int q = i - row * 48;
      int seg = q >> 4;   // 0: h[sender] 1: h[receiver] 2: h_edge
      int c4 = q & 15;
      const float* src;
      if (seg == 0)      src = HN + (size_t)iarr[row] * 64;
      else if (seg == 1) src = HN + (size_t)iarr[16 + row] * 64;
      else               src = HE + (size_t)(e0 + row) * 64;
      float4 v = *(const float4*)(src + (c4 << 2));
      *(float4*)(st + row * RS1 + (seg << 6) + (c4 << 2)) = v;
    }
    v8f acc[4];
    acc_bias_init<4>(acc, b1, lane);
    gemm_lds<4>(st, RS1, wf1, K1, lane, acc);      // 192 wmma
    c_to_lds<4>(ht, RSH, lane, acc, true);
    v8f acc2[4];
    acc_bias_init<4>(acc2, b2, lane);
    gemm_lds<4>(ht, RSH, wf2, 64, lane, acc2);     // 64 wmma
    int c = lane & 15, mb = (lane >> 4) << 3;
#pragma unroll
    for (int nt = 0; nt < 4; ++nt) {
      int n = nt * 16 + c;
#pragma unroll
      for (int r = 0; r < 8; ++r) {
        float v = silu(acc2[nt][r]);
        HE[(size_t)(e0 + mb + r) * 64 + n] = v;
        float* ap = AG + (size_t)iarr[16 + mb + r] * 64 + n;
        __hip_atomic_fetch_add(ap, v, __ATOMIC_RELAXED, __HIP_MEMORY_SCOPE_AGENT);
      }
    }
  }
}

// ---------------------------------------------------------------------------
// Node pass: state=[h_n|aggr] (128) -> L(128,64)->SiLU->L(64,64)  (in place)
// ---------------------------------------------------------------------------
__global__ void k_node_layer(float* __restrict__ HN, const float* __restrict__ AG,
                             const float* __restrict__ W1, const float* __restrict__ B1,
                             const float* __restrict__ W2, const float* __restrict__ B2) {
  extern __shared__ float smem[];
  const int K1 = 128, RS1 = 132, RSH = 68;
  float* wf1 = smem;                 // 128*64
  float* wf2 = wf1 + K1 * 64;
  float* b1 = wf2 + 64 * 64;
  float* b2 = b1 + 64;
  float* wbase = b2 + 64;
  int tid = threadIdx.x, lane = tid & 31, wid = tid >> 5;
  stage_weights(W1, wf1, K1, 64, 4, tid, TPB);
  stage_weights(W2, wf2, 64, 64, 4, tid, TPB);
  stage_bias(B1, b1, 64, tid, TPB);
  stage_bias(B2, b2, 64, tid, TPB);
  __syncthreads();
  float* st = wbase + wid * (16 * RS1 + 16 * RSH);
  float* ht = st + 16 * RS1;
  int gw = blockIdx.x * WPB + wid, nw = gridDim.x * WPB;
  const int ntiles = N_NODES / 16;
  for (int tile = gw; tile < ntiles; tile += nw) {
    int n0 = tile * 16;
    for (int i = lane; i < 512; i += 32) {  // 16 rows x 32 float4
      int row = i >> 5;
      int q = i & 31;
      int c4 = q & 15;
      const float* src = (q < 16) ? HN + (size_t)(n0 + row) * 64
                                  : AG + (size_t)(n0 + row) * 64;
      float4 v = *(const float4*)(src + (c4 << 2));
      *(float4*)(st + row * RS1 + ((q >> 4) << 6) + (c4 << 2)) = v;
    }
    v8f acc[4];
    acc_bias_init<4>(acc, b1, lane);
    gemm_lds<4>(st, RS1, wf1, K1, lane, acc);
    c_to_lds<4>(ht, RSH, lane, acc, true);
    v8f acc2[4];
    acc_bias_init<4>(acc2, b2, lane);
    gemm_lds<4>(ht, RSH, wf2, 64, lane, acc2);
    int c = lane & 15, mb = (lane >> 4) << 3;
#pragma unroll
    for (int nt = 0; nt < 4; ++nt) {
      int n = nt * 16 + c;
#pragma unroll
      for (int r = 0; r < 8; ++r)
        HN[(size_t)(n0 + mb + r) * 64 + n] = acc2[nt][r];
    }
  }
}

// ---------------------------------------------------------------------------
// Readout: [rows,64] -> L(64,64)->SiLU->L(64,3)  (W2 zero-padded to 16 cols)
// ---------------------------------------------------------------------------
__global__ void k_readout(const float* __restrict__ H, int ntiles,
                          const float* __restrict__ W1, const float* __restrict__ B1,
                          const float* __restrict__ W2, const float* __restrict__ B2,
                          float* __restrict__ OUT) {
  extern __shared__ float smem[];
  const int K1 = 64, RS1 = 68, RSH = 68;
  float* wf1 = smem;                 // 64*64
  float* wf2 = wf1 + K1 * 64;        // 64*16 (padded)
  float* b1 = wf2 + 64 * 16;
  float* b2 = b1 + 64;
  float* wbase = b2 + 64;
  int tid = threadIdx.x, lane = tid & 31, wid = tid >> 5;
  stage_weights(W1, wf1, K1, 64, 4, tid, TPB);
  stage_weights(W2, wf2, 64, 3, 1, tid, TPB);  // pad cols 3..15 with zero
  stage_bias(B1, b1, 64, tid, TPB);
  stage_bias(B2, b2, 3, tid, TPB);
  __syncthreads();
  float* st = wbase + wid * (16 * RS1 + 16 * RSH);
  float* ht = st + 16 * RS1;
  int gw = blockIdx.x * WPB + wid, nw = gridDim.x * WPB;
  for (int tile = gw; tile < ntiles; tile += nw) {
    int r0 = tile * 16;
    for (int i = lane; i < 256; i += 32) {  // 16 rows x 16 float4
      int row = i >> 4, c4 = i & 15;
      float4 v = *(const float4*)(H + (size_t)(r0 + row) * 64 + (c4 << 2));
      *(float4*)(st + row * RS1 + (c4 << 2)) = v;
    }
    v8f acc[4];
    acc_bias_init<4>(acc, b1, lane);
    gemm_lds<4>(st, RS1, wf1, K1, lane, acc);
    c_to_lds<4>(ht, RSH, lane, acc, true);
    v8f a2[1];
    acc_bias_init<1>(a2, b2, lane);
    gemm_lds<1>(ht, RSH, wf2, 64, lane, a2);
    int c = lane & 15, mb = (lane >> 4) << 3;
    if (c < 3) {
#pragma unroll
      for (int r = 0; r < 8; ++r)
        OUT[(size_t)(r0 + mb + r) * 3 + c] = a2[0][r];
    }
  }
}

// ---------------------------------------------------------------------------
// Host side
// ---------------------------------------------------------------------------
static inline int nblocks(int tiles) {
  int b = (tiles + WPB - 1) / WPB;
  return b < 960 ? b : 960;
}

extern "C" void kernel_launch(void* const* d_in, const int* in_sizes, int n_in,
                              void* d_out, int out_size, void* d_ws, size_t ws_size,
                              hipStream_t stream) {
  (void)in_sizes; (void)n_in; (void)out_size; (void)ws_size;

  const float* x_nodes = (const float*)d_in[0];
  const float* x_edges = (const float*)d_in[1];
  const int*   eidx    = (const int*)d_in[2];
  auto F = [&](int i) { return (const float*)d_in[i]; };
  // param leaf order (setup_inputs insertion order):
  // 3..6   embed_nodes  W1,b1,W2,b2
  // 7..10  embed_edges  W1,b1,W2,b2
  // 11..14 node_readout W1,b1,W2,b2
  // 15..18 edge_readout W1,b1,W2,b2
  // 19+8l  layer l: edge_net W1,b1,W2,b2 ; node_net W1,b1,W2,b2

  float* hn = (float*)d_ws;                           // [N,64]
  float* he = hn + (size_t)N_NODES * 64;              // [E,64]
  float* ag = he + (size_t)N_EDGES * 64;              // [N,64]

  const size_t shEmbed = (size_t)(16 * 64 + 64 * 64 + 128 +
                                  WPB * (16 * 20 + 16 * 68)) * 4;
  const size_t shEdge  = (size_t)(192 * 64 + 64 * 64 + 128 +
                                  WPB * (16 * 196 + 16 * 68 + 32)) * 4;
  const size_t shNode  = (size_t)(128 * 64 + 64 * 64 + 128 +
                                  WPB * (16 * 132 + 16 * 68)) * 4;
  const size_t shRead  = (size_t)(64 * 64 + 64 * 16 + 128 +
                                  WPB * (16 * 68 + 16 * 68)) * 4;

  // Allow >64KB dynamic LDS where needed (320KB/WGP available on CDNA5).
  (void)hipFuncSetAttribute((const void*)k_edge_layer,
                            hipFuncAttributeMaxDynamicSharedMemorySize, (int)shEdge);
  (void)hipFuncSetAttribute((const void*)k_node_layer,
                            hipFuncAttributeMaxDynamicSharedMemorySize, (int)shNode);

  const int tiles_n = N_NODES / 16;   // 3125
  const int tiles_e = N_EDGES / 16;   // 50000

  k_embed<<<nblocks(tiles_n), TPB, shEmbed, stream>>>(
      x_nodes, tiles_n, F(3), F(4), F(5), F(6), hn);
  k_embed<<<nblocks(tiles_e), TPB, shEmbed, stream>>>(
      x_edges, tiles_e, F(7), F(8), F(9), F(10), he);

  for (int l = 0; l < 4; ++l) {
    int b = 19 + 8 * l;
    hipMemsetAsync(ag, 0, (size_t)N_NODES * 64 * sizeof(float), stream);
    k_edge_layer<<<nblocks(tiles_e), TPB, shEdge, stream>>>(
        hn, he, eidx, ag, F(b + 0), F(b + 1), F(b + 2), F(b + 3));
    k_node_layer<<<nblocks(tiles_n), TPB, shNode, stream>>>(
        hn, ag, F(b + 4), F(b + 5), F(b + 6), F(b + 7));
  }

  float* out_nodes = (float*)d_out;                   // [N,3]
  float* out_edges = out_nodes + (size_t)N_NODES * 3; // [E,3]
  k_readout<<<nblocks(tiles_n), TPB, shRead, stream>>>(
      hn, tiles_n, F(11), F(12), F(13), F(14), out_nodes);
  k_readout<<<nblocks(tiles_e), TPB, shRead, stream>>>(
      he, tiles_e, F(15), F(16), F(17), F(18), out_edges);
}